// MultiHeadAttentionWithDropoutAndMask_74062416052664
// MI455X (gfx1250) — compile-verified
//
#include <hip/hip_runtime.h>
#include <hip/hip_bf16.h>

typedef __attribute__((ext_vector_type(2))) float v2f;
typedef __attribute__((ext_vector_type(8))) float v8f;

constexpr int BATCH = 8;
constexpr int SEQ   = 4096;
constexpr int DIM   = 1024;
constexpr int HEADS = 16;
constexpr float SCALE = 0.125f;   // 1/sqrt(64)

// ---------------------------------------------------------------------------
// Kernel 1: logits[b,s,h] = (x[b,s,:] . W[:,h]) * SCALE, masked to -inf.
// One wave (32 threads) per 16-row s-tile; V_WMMA_F32_16X16X4_F32, K=1024.
// A (16x4): lane m holds row m, K = k0+2*half .. +1  -> contiguous float2
// B (4x16): lane n holds col n, VGPR i = row k0+i+2*half
// ---------------------------------------------------------------------------
__global__ __launch_bounds__(32)
void k_logits(const float* __restrict__ x, const float* __restrict__ W,
              const int* __restrict__ mask, float* __restrict__ logits)
{
    const int lane = threadIdx.x;
    const int m    = lane & 15;     // A row (s within tile) / B col (head)
    const int half = lane >> 4;
    const int tile = blockIdx.x;                  // 0 .. BATCH*SEQ/16-1
    const int b    = tile >> 8;                   // tile / (SEQ/16)
    const int s0   = (tile & 255) << 4;

    const float* xrow = x + ((size_t)b * SEQ + (s0 + m)) * DIM;
    v8f c = {};

    #pragma unroll 4
    for (int k0 = 0; k0 < DIM; k0 += 4) {
        const int ka = k0 + 2 * half;
        v2f a = *(const v2f*)(xrow + ka);          // x[b, s0+m, ka..ka+1]
        v2f bf;
        bf.x = W[(size_t)(ka + 0) * HEADS + m];    // W[ka,   m]
        bf.y = W[(size_t)(ka + 1) * HEADS + m];    // W[ka+1, m]
        c = __builtin_amdgcn_wmma_f32_16x16x4_f32(
                false, a, false, bf, (short)0, c, false, false);
    }

    #pragma unroll
    for (int r = 0; r < 8; ++r) {
        const int s = s0 + r + 8 * half;           // D layout: M = r + 8*half
        float v = c[r] * SCALE;
        if (mask[(size_t)b * SEQ + s] == 0) v = -__builtin_inff();
        logits[((size_t)b * SEQ + s) * HEADS + m] = v;
    }
}

// ---------------------------------------------------------------------------
// Kernel 2: softmax over the sequence axis, in place.
// One 256-thread block per (b,h); each thread owns 16 elements in registers.
// ---------------------------------------------------------------------------
__global__ __launch_bounds__(256)
void k_softmax(float* __restrict__ logits)
{
    __shared__ float red[256];
    const int bh = blockIdx.x;                    // 0 .. BATCH*HEADS-1
    const int b  = bh >> 4;
    const int h  = bh & 15;
    const int t  = threadIdx.x;
    float* base = logits + (size_t)b * SEQ * HEADS + h;

    float vals[16];
    float mx = -__builtin_inff();
    #pragma unroll
    for (int i = 0; i < 16; ++i) {
        float v = base[(size_t)(i * 256 + t) * HEADS];
        vals[i] = v;
        mx = fmaxf(mx, v);
    }
    red[t] = mx; __syncthreads();
    for (int off = 128; off > 0; off >>= 1) {
        if (t < off) red[t] = fmaxf(red[t], red[t + off]);
        __syncthreads();
    }
    mx = red[0];
    __syncthreads();

    float sum = 0.0f;
    #pragma unroll
    for (int i = 0; i < 16; ++i) {
        float e = __expf(vals[i] - mx);            // exp(-inf - mx) = 0
        vals[i] = e;
        sum += e;
    }
    red[t] = sum; __syncthreads();
    for (int off = 128; off > 0; off >>= 1) {
        if (t < off) red[t] += red[t + off];
        __syncthreads();
    }
    const float rinv = 1.0f / red[0];

    #pragma unroll
    for (int i = 0; i < 16; ++i)
        base[(size_t)(i * 256 + t) * HEADS] = vals[i] * rinv;
}

// ---------------------------------------------------------------------------
// Kernel 3: pooled[b,h,d] = sum_s attn[b,s,h] * x[b,s,d]
// One wave per (b, 16-wide d-tile): M=16 heads, N=16 d-cols, K=4096 seq.
// A (16x4): lane m = head m, K pair by half (attn strided by HEADS)
// B (4x16): lane n = col d0+n, VGPR i = seq row ka+i (coalesced 64B/halfwave)
// ---------------------------------------------------------------------------
__global__ __launch_bounds__(32)
void k_pool(const float* __restrict__ x, const float* __restrict__ attn,
            float* __restrict__ out)
{
    const int lane = threadIdx.x;
    const int n    = lane & 15;     // head for A frag, d-col for B frag
    const int half = lane >> 4;
    const int tile = blockIdx.x;                  // 0 .. BATCH*DIM/16-1
    const int b    = tile >> 6;                   // tile / (DIM/16)
    const int d0   = (tile & 63) << 4;

    const float* xb = x    + (size_t)b * SEQ * DIM   + d0 + n;
    const float* ab = attn + (size_t)b * SEQ * HEADS + n;
    v8f c = {};

    #pragma unroll 4
    for (int k0 = 0; k0 < SEQ; k0 += 4) {
        const int ka = k0 + 2 * half;
        v2f a;
        a.x = ab[(size_t)(ka + 0) * HEADS];        // attn[b, ka,   n]
        a.y = ab[(size_t)(ka + 1) * HEADS];        // attn[b, ka+1, n]
        v2f bf;
        bf.x = xb[(size_t)(ka + 0) * DIM];         // x[b, ka,   d0+n]
        bf.y = xb[(size_t)(ka + 1) * DIM];         // x[b, ka+1, d0+n]
        c = __builtin_amdgcn_wmma_f32_16x16x4_f32(
                false, a, false, bf, (short)0, c, false, false);
    }

    #pragma unroll
    for (int r = 0; r < 8; ++r) {
        const int hrow = r + 8 * half;             // D layout: M = r + 8*half
        out[(size_t)b * (HEADS * DIM) + (size_t)hrow * DIM + d0 + n] = c[r];
    }
}

// ---------------------------------------------------------------------------
extern "C" void kernel_launch(void* const* d_in, const int* in_sizes, int n_in,
                              void* d_out, int out_size, void* d_ws, size_t ws_size,
                              hipStream_t stream)
{
    const float* x    = (const float*)d_in[0];   // [8, 4096, 1024] f32
    const float* W    = (const float*)d_in[1];   // [1024, 16]      f32
    const int*   mask = (const int*)  d_in[2];   // [8, 4096]       i32
    float* out    = (float*)d_out;               // [8, 16384]      f32
    float* logits = (float*)d_ws;                // B*S*H floats = 2 MB scratch

    k_logits <<<dim3(BATCH * SEQ / 16), dim3(32),  0, stream>>>(x, W, mask, logits);
    k_softmax<<<dim3(BATCH * HEADS),    dim3(256), 0, stream>>>(logits);
    k_pool   <<<dim3(BATCH * DIM / 16), dim3(32),  0, stream>>>(x, logits, out);
}